// GCN_4526895529990
// MI455X (gfx1250) — compile-verified
//
#include <hip/hip_runtime.h>
#include <hip/hip_bf16.h>

typedef float v2f __attribute__((ext_vector_type(2)));
typedef float v8f __attribute__((ext_vector_type(8)));

// ---------------------------------------------------------------------------
// Degree / normalization
// ---------------------------------------------------------------------------
__global__ void k_deg_init(int* __restrict__ deg, int n) {
    int i = blockIdx.x * blockDim.x + threadIdx.x;
    if (i < n) deg[i] = 1;  // self-loop contributes 1
}

__global__ void k_deg_count(const int* __restrict__ dst, int* __restrict__ deg, int e) {
    int i = blockIdx.x * blockDim.x + threadIdx.x;
    if (i < e) atomicAdd(&deg[dst[i]], 1);  // L2-resident u32 atomic, no return
}

__global__ void k_dinv(float* __restrict__ dinv, int n) {
    int i = blockIdx.x * blockDim.x + threadIdx.x;
    if (i < n) {
        int d = ((const int*)dinv)[i];         // in-place int -> float rsqrt
        dinv[i] = rsqrtf((float)d);            // d >= 1 always (self-loop)
    }
}

// ---------------------------------------------------------------------------
// Layer 1 GEMM: h1[n,8] = x[n,3] @ W1[3,8]  via V_WMMA_F32_16X16X4_F32
// One wave -> 16 nodes. A: 16x4 (K=3 padded), B: 4x16 (N=8 padded).
// Branch-free operand setup (clamped loads + cndmask selects); full-tile
// fast path emits 8 unconditional stores off one base pointer.
// ---------------------------------------------------------------------------
__global__ void k_gemm1(const float* __restrict__ x, const float* __restrict__ W1,
                        float* __restrict__ h1, int n) {
    int lane = threadIdx.x & 31;
    int wave = threadIdx.x >> 5;
    int base = (blockIdx.x * (blockDim.x >> 5) + wave) * 16;
    if (base >= n) return;  // wave-uniform: EXEC stays all-1s for WMMA

    int  m   = lane & 15;
    bool hi  = lane >= 16;
    int node = base + m;
    if (node >= n) node = n - 1;  // clamp loads; stores guarded below
    const float* xr = x + (size_t)node * 3;

    // A 16x4: lanes 0-15 -> (K0,K1); lanes 16-31 -> (K2, 0-pad). Branchless:
    float ax = xr[hi ? 2 : 0];     // one load, selected offset
    float x1 = xr[1];              // unconditional load, masked below
    v2f a;
    a.x = ax;
    a.y = hi ? 0.0f : x1;

    // B 4x16: row-major W1[k][m], m = lane%16; pad m>=8 and K=3 with zero.
    int  mc    = (m < 8) ? m : 7;     // clamped column: loads always in-bounds
    bool valid = (m < 8);
    float w0 = W1[0 * 8 + mc];
    float w1 = W1[1 * 8 + mc];
    float w2 = W1[2 * 8 + mc];
    v2f b;
    b.x = valid ? (hi ? w2 : w0) : 0.0f;
    b.y = (valid && !hi) ? w1 : 0.0f;

    v8f c = {};
    c = __builtin_amdgcn_wmma_f32_16x16x4_f32(false, a, false, b, (short)0, c,
                                              false, false);

    // D: VGPR r, lane L -> row base + r + (hi?8:0), col m
    if (valid) {
        float* hp = h1 + (size_t)(base + (hi ? 8 : 0)) * 8 + m;
        if (base + 16 <= n) {          // full tile (the common case)
            #pragma unroll
            for (int r = 0; r < 8; ++r) hp[(size_t)r * 8] = c[r];
        } else {                       // tail tile
            #pragma unroll
            for (int r = 0; r < 8; ++r) {
                int row = base + r + (hi ? 8 : 0);
                if (row < n) h1[(size_t)row * 8 + m] = c[r];
            }
        }
    }
}

// ---------------------------------------------------------------------------
// agg1[v][j] = b1[j] + h1[v][j] * dinv[v]^2   (bias + self-loop message)
// ---------------------------------------------------------------------------
__global__ void k_init1(const float* __restrict__ h1, const float* __restrict__ dinv,
                        const float* __restrict__ b1, float* __restrict__ agg1, int n8) {
    int i = blockIdx.x * blockDim.x + threadIdx.x;
    if (i >= n8) return;
    int   v  = i >> 3, j = i & 7;
    float di = dinv[v];
    agg1[i] = b1[j] + h1[i] * di * di;
}

// ---------------------------------------------------------------------------
// Edge scatter layer 1: agg1[d] += h1[s] * dinv[s]*dinv[d]
// h1/agg1/dinv are L2-resident (68MB < 192MB L2); edges stream at HBM rate.
// 8 no-return global_atomic_add_f32 per edge = the true workload bottleneck.
// ---------------------------------------------------------------------------
__global__ void k_edge1(const int* __restrict__ src, const int* __restrict__ dst,
                        const float* __restrict__ dinv, const float* __restrict__ h1,
                        float* __restrict__ agg1, int e) {
    int i = blockIdx.x * blockDim.x + threadIdx.x;
    if (i >= e) return;
    int s = src[i], d = dst[i];
    float nrm = dinv[s] * dinv[d];
    const float4* hp = (const float4*)(h1 + (size_t)s * 8);  // 32B-aligned B128 x2
    float4 m0 = hp[0], m1 = hp[1];
    float* o = agg1 + (size_t)d * 8;
    unsafeAtomicAdd(o + 0, m0.x * nrm);
    unsafeAtomicAdd(o + 1, m0.y * nrm);
    unsafeAtomicAdd(o + 2, m0.z * nrm);
    unsafeAtomicAdd(o + 3, m0.w * nrm);
    unsafeAtomicAdd(o + 4, m1.x * nrm);
    unsafeAtomicAdd(o + 5, m1.y * nrm);
    unsafeAtomicAdd(o + 6, m1.z * nrm);
    unsafeAtomicAdd(o + 7, m1.w * nrm);
}

// ---------------------------------------------------------------------------
// Layer 2 GEMM: h2[n,2] = relu(agg1[n,8]) @ W2[8,2]
// Two chained 16x16x4 WMMAs (K split 0..3 / 4..7), ReLU fused into A loads.
// ---------------------------------------------------------------------------
__global__ void k_gemm2(const float* __restrict__ agg1, const float* __restrict__ W2,
                        float* __restrict__ h2, int n) {
    int lane = threadIdx.x & 31;
    int wave = threadIdx.x >> 5;
    int base = (blockIdx.x * (blockDim.x >> 5) + wave) * 16;
    if (base >= n) return;  // wave-uniform

    int  m   = lane & 15;
    bool hi  = lane >= 16;
    int node = base + m;
    if (node >= n) node = n - 1;
    const float* r0 = agg1 + (size_t)node * 8;
    int k0 = hi ? 2 : 0;

    // Branchless A loads (computed K offset) with fused ReLU
    v2f a0, a1;
    a0.x = fmaxf(r0[k0],     0.0f); a0.y = fmaxf(r0[k0 + 1],     0.0f);
    a1.x = fmaxf(r0[4 + k0], 0.0f); a1.y = fmaxf(r0[4 + k0 + 1], 0.0f);

    // W2 row-major [8][2]; column m, valid for m < 2. Clamp + select.
    int  mc    = (m < 2) ? m : 1;
    bool valid = (m < 2);
    float wa = W2[(k0)     * 2 + mc];
    float wb = W2[(k0 + 1) * 2 + mc];
    float wc = W2[(k0 + 4) * 2 + mc];
    float wd = W2[(k0 + 5) * 2 + mc];
    v2f b0, b1v;
    b0.x  = valid ? wa : 0.0f;
    b0.y  = valid ? wb : 0.0f;
    b1v.x = valid ? wc : 0.0f;
    b1v.y = valid ? wd : 0.0f;

    v8f c = {};
    c = __builtin_amdgcn_wmma_f32_16x16x4_f32(false, a0, false, b0, (short)0, c,
                                              false, false);
    c = __builtin_amdgcn_wmma_f32_16x16x4_f32(false, a1, false, b1v, (short)0, c,
                                              false, false);

    if (valid) {
        float* hp = h2 + (size_t)(base + (hi ? 8 : 0)) * 2 + m;
        if (base + 16 <= n) {          // full tile fast path
            #pragma unroll
            for (int r = 0; r < 8; ++r) hp[(size_t)r * 2] = c[r];
        } else {
            #pragma unroll
            for (int r = 0; r < 8; ++r) {
                int row = base + r + (hi ? 8 : 0);
                if (row < n) h2[(size_t)row * 2 + m] = c[r];
            }
        }
    }
}

// ---------------------------------------------------------------------------
// out[v][j] = b2[j] + h2[v][j] * dinv[v]^2   (bias + self-loop, layer 2)
// ---------------------------------------------------------------------------
__global__ void k_init2(const float* __restrict__ h2, const float* __restrict__ dinv,
                        const float* __restrict__ b2, float* __restrict__ out, int n2) {
    int i = blockIdx.x * blockDim.x + threadIdx.x;
    if (i >= n2) return;
    int   v  = i >> 1, j = i & 1;
    float di = dinv[v];
    out[i] = b2[j] + h2[i] * di * di;
}

// ---------------------------------------------------------------------------
// Edge scatter layer 2: out[d] += h2[s] * dinv[s]*dinv[d]   (2 feats)
// ---------------------------------------------------------------------------
__global__ void k_edge2(const int* __restrict__ src, const int* __restrict__ dst,
                        const float* __restrict__ dinv, const float* __restrict__ h2,
                        float* __restrict__ out, int e) {
    int i = blockIdx.x * blockDim.x + threadIdx.x;
    if (i >= e) return;
    int s = src[i], d = dst[i];
    float nrm = dinv[s] * dinv[d];
    const float2* hp = (const float2*)(h2 + (size_t)s * 2);  // 8B-aligned
    float2 m0 = hp[0];
    float* o = out + (size_t)d * 2;
    unsafeAtomicAdd(o + 0, m0.x * nrm);
    unsafeAtomicAdd(o + 1, m0.y * nrm);
}

// ---------------------------------------------------------------------------
// Launcher
// ---------------------------------------------------------------------------
extern "C" void kernel_launch(void* const* d_in, const int* in_sizes, int n_in,
                              void* d_out, int out_size, void* d_ws, size_t ws_size,
                              hipStream_t stream) {
    const float* x  = (const float*)d_in[0];
    const int*   ei = (const int*)d_in[1];
    const float* W1 = (const float*)d_in[2];
    const float* b1 = (const float*)d_in[3];
    const float* W2 = (const float*)d_in[4];
    const float* b2 = (const float*)d_in[5];
    float* out = (float*)d_out;

    int n = in_sizes[0] / 3;   // 1,000,000 nodes
    int e = in_sizes[1] / 2;   // 10,000,000 edges
    const int* src = ei;
    const int* dst = ei + e;

    // workspace layout (68 MB): [dinv: n f32][h1: 8n f32][agg1: 8n f32]; h2 reuses h1
    float* dinv = (float*)d_ws;
    float* h1   = dinv + n;
    float* agg1 = h1 + (size_t)n * 8;
    float* h2   = h1;  // h1 dead after k_edge1; 2n floats fit in its 8n slot

    const int T = 256;                       // 8 waves/block (wave32)
    int gN  = (n + T - 1) / T;
    int gE  = (e + T - 1) / T;
    int gN8 = (n * 8 + T - 1) / T;
    int gN2 = (n * 2 + T - 1) / T;
    int gMM = (n + 127) / 128;               // 8 waves * 16 nodes per block

    k_deg_init <<<gN,  T, 0, stream>>>((int*)dinv, n);
    k_deg_count<<<gE,  T, 0, stream>>>(dst, (int*)dinv, e);
    k_dinv     <<<gN,  T, 0, stream>>>(dinv, n);
    k_gemm1    <<<gMM, T, 0, stream>>>(x, W1, h1, n);
    k_init1    <<<gN8, T, 0, stream>>>(h1, dinv, b1, agg1, n * 8);
    k_edge1    <<<gE,  T, 0, stream>>>(src, dst, dinv, h1, agg1, e);
    k_gemm2    <<<gMM, T, 0, stream>>>(agg1, W2, h2, n);
    k_init2    <<<gN2, T, 0, stream>>>(h2, dinv, b2, out, n * 2);
    k_edge2    <<<gE,  T, 0, stream>>>(src, dst, dinv, h2, out, e);
}